// SplineGCN_7859790152276
// MI455X (gfx1250) — compile-verified
//
#include <hip/hip_runtime.h>
#include <hip/hip_bf16.h>

typedef float v2f __attribute__((ext_vector_type(2)));
typedef float v8f __attribute__((ext_vector_type(8)));

// ---------------------------------------------------------------------------
// f32 WMMA GEMM: C[M x (NT*16)] = A[M x (KSTEPS*4)] * B, row-major.
// Preconditions (arranged by the host):  M % 16 == 0, N = NT*16 exactly
// (layer-2 weights are zero-padded 40 -> 48), K = KSTEPS*4 exactly.
// B is staged into LDS in WMMA fragment order [kstep][lane][tile] (float2),
// so the K-loop is: 1x global_load_b64 (A) + contiguous LDS reads + NT WMMAs,
// with zero predication (EXEC all-ones throughout).
// ---------------------------------------------------------------------------
template <int NT, int KSTEPS>
__global__ void wmma_gemm_f32(const float* __restrict__ A, int lda,
                              const float* __restrict__ B, int ldb,
                              float* __restrict__ C, int ldc, int M)
{
    __shared__ v2f bfrag[KSTEPS * 32 * 4];   // fixed tile stride 4

    const int lane = (int)threadIdx.x & 31;
    const int lrow = lane & 15;
    const int half = lane >> 4;              // 0: K pair {0,1}; 1: {2,3}

    // ---- cooperative stage of B into LDS (fragment order) ----
    for (int idx = threadIdx.x; idx < KSTEPS * 128; idx += blockDim.x) {
        const int step = idx >> 7;           // / 128
        const int r    = idx & 127;
        const int ln   = r >> 2;
        const int tile = r & 3;
        v2f val = {0.0f, 0.0f};
        if (tile < NT) {
            const int ka = step * 4 + ((ln >> 4) << 1);
            const int n  = tile * 16 + (ln & 15);
            val.x = B[(size_t)ka * ldb + n];
            val.y = B[(size_t)(ka + 1) * ldb + n];
        }
        bfrag[idx] = val;
    }
    __syncthreads();

    const int wave = blockIdx.x * (blockDim.x >> 5) + ((int)threadIdx.x >> 5);
    const int row0 = wave << 4;
    if (row0 >= M) return;                   // wave-uniform, after the barrier

    const int m = row0 + lrow;
    const float* __restrict__ Arow = A + (size_t)m * lda;

    v8f acc[NT];
    const v8f vzero = {0.f,0.f,0.f,0.f,0.f,0.f,0.f,0.f};
#pragma unroll
    for (int t = 0; t < NT; ++t) acc[t] = vzero;

#pragma unroll 4
    for (int s = 0; s < KSTEPS; ++s) {
        const int ka = s * 4 + (half << 1);
        const v2f a = *(const v2f*)(Arow + ka);          // global_load_b64
        const v2f* __restrict__ p = &bfrag[(s * 32 + lane) * 4];
#pragma unroll
        for (int t = 0; t < NT; ++t) {
            const v2f b = p[t];                          // contiguous LDS
            acc[t] = __builtin_amdgcn_wmma_f32_16x16x4_f32(
                false, a, false, b, (short)0, acc[t], false, false);
        }
    }

    const int mbase = row0 + (half << 3);
#pragma unroll
    for (int t = 0; t < NT; ++t) {
        const int n = (t << 4) + lrow;
#pragma unroll
        for (int v = 0; v < 8; ++v)
            C[(size_t)(mbase + v) * ldc + n] = acc[t][v];
    }
}

__global__ void zero_f32(float* __restrict__ p, long long n)
{
    const long long stride = (long long)gridDim.x * blockDim.x;
    for (long long i = blockIdx.x * (long long)blockDim.x + threadIdx.x;
         i < n; i += stride)
        p[i] = 0.0f;
}

// Zero-pad B[K x N] (row-major) into Bp[K x Np]
__global__ void pad_b(const float* __restrict__ B, float* __restrict__ Bp,
                      int K, int N, int Np)
{
    const int i = blockIdx.x * blockDim.x + threadIdx.x;
    if (i < K * Np) {
        const int k = i / Np, n = i - k * Np;
        Bp[i] = (n < N) ? B[k * N + n] : 0.0f;
    }
}

__global__ void deg_count(const int* __restrict__ dst, float* __restrict__ deg, int E)
{
    const int i = blockIdx.x * blockDim.x + threadIdx.x;
    if (i < E) unsafeAtomicAdd(&deg[dst[i]], 1.0f);
}

// Per-(edge, 4-feature) thread: acc[dst,f..f+3] += (1-v)*Y0[src] + v*Y1[src]
// Y row n: [Y0 at col 0 | Y1 at col OFF1 | root ...], row stride ldy.
template <int F4, int F, int OFF1>
__global__ void edge_agg4(const int* __restrict__ src, const int* __restrict__ dst,
                          const float* __restrict__ attr,
                          const float* __restrict__ Y, int ldy,
                          float* __restrict__ acc, long long total)
{
    const long long stride = (long long)gridDim.x * blockDim.x;
    for (long long i = blockIdx.x * (long long)blockDim.x + threadIdx.x;
         i < total; i += stride) {
        const int e = (int)(i / F4);                 // F4 is constexpr -> mul-shift
        const int f = ((int)(i - (long long)e * F4)) << 2;
        const float v  = attr[e];
        const float w0 = 1.0f - v;
        const int s = src[e];
        const int d = dst[e];
        const float* __restrict__ yr = Y + (size_t)s * ldy;
        const float4 a0 = *(const float4*)(yr + f);           // global_load_b128
        const float4 a1 = *(const float4*)(yr + OFF1 + f);
        float* __restrict__ ap = acc + (size_t)d * F + f;
        unsafeAtomicAdd(ap + 0, w0 * a0.x + v * a1.x);
        unsafeAtomicAdd(ap + 1, w0 * a0.y + v * a1.y);
        unsafeAtomicAdd(ap + 2, w0 * a0.z + v * a1.z);
        unsafeAtomicAdd(ap + 3, w0 * a0.w + v * a1.w);
    }
}

// out[n,f..f+3] = act( acc/max(deg,1) + Y[n, rootoff+f] + bias[f] )
template <int F4, int F>
__global__ void finalize4(const float* __restrict__ acc,
                          const float* __restrict__ Y, int ldy, int rootoff,
                          const float* __restrict__ deg,
                          const float* __restrict__ bias,
                          float* __restrict__ out,
                          long long total, int apply_elu)
{
    const long long stride = (long long)gridDim.x * blockDim.x;
    for (long long i = blockIdx.x * (long long)blockDim.x + threadIdx.x;
         i < total; i += stride) {
        const int n = (int)(i / F4);
        const int f = ((int)(i - (long long)n * F4)) << 2;
        const float dinv = 1.0f / fmaxf(deg[n], 1.0f);
        const float4 ac = *(const float4*)(acc + (size_t)n * F + f);
        const float4 rt = *(const float4*)(Y + (size_t)n * ldy + rootoff + f);
        const float4 bs = *(const float4*)(bias + f);
        float4 val;
        val.x = ac.x * dinv + rt.x + bs.x;
        val.y = ac.y * dinv + rt.y + bs.y;
        val.z = ac.z * dinv + rt.z + bs.z;
        val.w = ac.w * dinv + rt.w + bs.w;
        if (apply_elu) {
            val.x = (val.x > 0.f) ? val.x : expm1f(val.x);
            val.y = (val.y > 0.f) ? val.y : expm1f(val.y);
            val.z = (val.z > 0.f) ? val.z : expm1f(val.z);
            val.w = (val.w > 0.f) ? val.w : expm1f(val.w);
        }
        *(float4*)(out + i * 4) = val;
    }
}

static inline int iminh(int a, int b) { return a < b ? a : b; }
static inline int blocksFor(long long n, int tb) { return (int)((n + tb - 1) / tb); }

extern "C" void kernel_launch(void* const* d_in, const int* in_sizes, int n_in,
                              void* d_out, int out_size, void* d_ws, size_t ws_size,
                              hipStream_t stream)
{
    const float* x     = (const float*)d_in[0];
    const int*   eidx  = (const int*)  d_in[1];
    const float* attr  = (const float*)d_in[2];
    const float* W1    = (const float*)d_in[3];   // [2,128,64]
    const float* root1 = (const float*)d_in[4];   // [128,64]
    const float* bias1 = (const float*)d_in[5];   // [64]
    const float* W2    = (const float*)d_in[6];   // [2,64,40]
    const float* root2 = (const float*)d_in[7];   // [64,40]
    const float* bias2 = (const float*)d_in[8];   // [40]
    float* out = (float*)d_out;

    const int FIN = 128, HID = 64, NCLS = 40, NPAD = 48;
    const int Nn = in_sizes[0] / FIN;   // 50000 (multiple of 16)
    const int E  = in_sizes[1] / 2;     // 800000
    const int* src = eidx;
    const int* dst = eidx + E;

    // Workspace (floats):
    //   Ycat : Nn*192  (layer1); layer2 Zcat (Nn*144) aliases it
    //   deg  : Nn
    //   acc1 : Nn*64   (layer2 acc2 aliases it)
    //   h    : Nn*64
    //   bpad : 3 * 64*48  (zero-padded layer-2 weights)
    float* ws   = (float*)d_ws;
    float* Ycat = ws;
    float* deg  = ws   + (size_t)Nn * 192;
    float* acc1 = deg  + Nn;
    float* h    = acc1 + (size_t)Nn * HID;
    float* bpad = h    + (size_t)Nn * HID;

    const int TB = 256;
    const int strips  = (Nn + 15) / 16;                      // 3125
    const int gblocks = (strips + (TB / 32) - 1) / (TB / 32);

    // ---- zero deg + acc1 (contiguous) ----
    {
        long long n = (long long)Nn * (1 + HID);
        zero_f32<<<iminh(blocksFor(n, TB), 16384), TB, 0, stream>>>(deg, n);
    }
    // ---- degrees (shared by both layers) ----
    deg_count<<<blocksFor(E, TB), TB, 0, stream>>>(dst, deg, E);
    // ---- pad layer-2 weights 40 -> 48 cols ----
    {
        const int pe = HID * NPAD;
        pad_b<<<blocksFor(pe, TB), TB, 0, stream>>>(W2,               bpad,          HID, NCLS, NPAD);
        pad_b<<<blocksFor(pe, TB), TB, 0, stream>>>(W2 + HID * NCLS, bpad + pe,     HID, NCLS, NPAD);
        pad_b<<<blocksFor(pe, TB), TB, 0, stream>>>(root2,            bpad + 2 * pe, HID, NCLS, NPAD);
    }

    // ---- layer 1 GEMMs: Ycat = X @ [W1[0] | W1[1] | root1], N=64, K=128 ----
    wmma_gemm_f32<4, 32><<<gblocks, TB, 0, stream>>>(x, FIN, W1,                   HID, Ycat,       192, Nn);
    wmma_gemm_f32<4, 32><<<gblocks, TB, 0, stream>>>(x, FIN, W1 + (size_t)FIN*HID, HID, Ycat + 64,  192, Nn);
    wmma_gemm_f32<4, 32><<<gblocks, TB, 0, stream>>>(x, FIN, root1,                HID, Ycat + 128, 192, Nn);

    // ---- layer 1 edge aggregation (float4, 16 quads per edge) ----
    {
        long long total = (long long)E * (HID / 4);
        edge_agg4<16, 64, 64><<<iminh(blocksFor(total, TB), 262144), TB, 0, stream>>>(
            src, dst, attr, Ycat, 192, acc1, total);
    }
    // ---- layer 1 finalize: h = elu(acc/deg + X@root1 + bias1) ----
    {
        long long total = (long long)Nn * (HID / 4);
        finalize4<16, 64><<<iminh(blocksFor(total, TB), 65536), TB, 0, stream>>>(
            acc1, Ycat, 192, 128, deg, bias1, h, total, 1);
    }

    // ---- layer 2 (Zcat = h @ [W2[0]p | W2[1]p | root2p], N=48 padded, K=64) ----
    float* Zcat = ws;          // Nn*144, aliases retired Ycat
    float* acc2 = acc1;        // Nn*40, aliases retired acc1
    {
        long long n = (long long)Nn * NCLS;
        zero_f32<<<iminh(blocksFor(n, TB), 16384), TB, 0, stream>>>(acc2, n);
    }
    {
        const int pe = HID * NPAD;
        wmma_gemm_f32<3, 16><<<gblocks, TB, 0, stream>>>(h, HID, bpad,          NPAD, Zcat,      144, Nn);
        wmma_gemm_f32<3, 16><<<gblocks, TB, 0, stream>>>(h, HID, bpad + pe,     NPAD, Zcat + 48, 144, Nn);
        wmma_gemm_f32<3, 16><<<gblocks, TB, 0, stream>>>(h, HID, bpad + 2 * pe, NPAD, Zcat + 96, 144, Nn);
    }
    // ---- layer 2 edge aggregation (10 quads per edge) ----
    {
        long long total = (long long)E * (NCLS / 4);
        edge_agg4<10, 40, 48><<<iminh(blocksFor(total, TB), 262144), TB, 0, stream>>>(
            src, dst, attr, Zcat, 144, acc2, total);
    }
    // ---- layer 2 finalize -> d_out ----
    {
        long long total = (long long)Nn * (NCLS / 4);
        finalize4<10, 40><<<iminh(blocksFor(total, TB), 65536), TB, 0, stream>>>(
            acc2, Zcat, 144, 96, deg, bias2, out, total, 0);
    }
}